// Closing_76613626626392
// MI455X (gfx1250) — compile-verified
//
#include <hip/hip_runtime.h>
#include <stdint.h>

// Problem constants (match reference)
#define B_    4
#define CIN   3
#define COUT  16
#define H_    256
#define W_    256
#define K_    5
#define TILE  32
#define DT    (TILE + 4)   // 36: dilated tile incl. erosion halo (+/-2)
#define XT    (TILE + 8)   // 40: x tile incl. dilation+erosion halo (+/-4)
#define DSTR  40           // dil row stride (floats): 160B -> 16B-aligned float4 rows
#define NWD   (CIN * K_ * K_)   // 75
#define NWE   (K_ * K_)         // 25

typedef float v4f __attribute__((ext_vector_type(4)));

// Generic pointer -> 32-bit LDS byte offset (addrspacecast to LDS, truncate).
__device__ __forceinline__ uint32_t lds_off(void* p) {
    return (uint32_t)(uintptr_t)(__attribute__((address_space(3))) void*)p;
}

// CDNA5 async DMA: global -> LDS, 4 bytes per active lane, tracked by ASYNCcnt.
__device__ __forceinline__ void async_copy_b32(uint32_t lds_byte_addr, const float* gaddr) {
    asm volatile("global_load_async_to_lds_b32 %0, %1, off"
                 :: "v"(lds_byte_addr), "v"(gaddr)
                 : "memory");
}

__device__ __forceinline__ void wait_async_all() {
    asm volatile("s_wait_asynccnt 0" ::: "memory");
}

__device__ __forceinline__ int clampi(int v, int lo, int hi) {
    return v < lo ? lo : (v > hi ? hi : v);
}

__global__ __launch_bounds__(256) void closing_fused_kernel(
    const float* __restrict__ x,    // (B, CIN, H, W)
    const float* __restrict__ wd,   // (COUT, CIN, K, K)
    const float* __restrict__ we,   // (COUT, K, K)
    float* __restrict__ out)        // (B, COUT, H, W)
{
    __shared__ float xs[CIN][XT][XT];       // 19200 B
    __shared__ float dil[DT][DSTR];         //  5760 B (16B-aligned rows)
    __shared__ float wds[NWD];              //   300 B
    __shared__ float wes[NWE];              //   100 B

    const int t   = threadIdx.x;
    const int tpx = W_ / TILE;                 // 8 tiles per row
    const int tix = blockIdx.x % tpx;
    const int tiy = blockIdx.x / tpx;
    const int o   = blockIdx.y;
    const int b   = blockIdx.z;
    const int x0  = tix * TILE;
    const int y0  = tiy * TILE;
    // Interior tile: no edge clamp anywhere in the 40x40 x-window.
    const bool interior = (tix >= 1) && (tix <= tpx - 2) && (tiy >= 1) && (tiy <= tpx - 2);

    // ---- stage weights into LDS (broadcast reads later, conflict-free) ----
    if (t < NWD) {
        wds[t] = wd[o * NWD + t];
    } else if (t < NWD + NWE) {
        wes[t - NWD] = we[o * NWE + (t - NWD)];
    }

    // ---- async DMA the x tile (3 x 40 x 40) with edge-clamped gather ----
    const float* xb = x + (size_t)b * CIN * H_ * W_;
    const int NX = CIN * XT * XT;              // 4800 elements
    for (int e = t; e < NX; e += 256) {
        int c   = e / (XT * XT);
        int rem = e - c * (XT * XT);
        int r   = rem / XT;
        int col = rem - r * XT;
        int gr  = clampi(y0 - 4 + r,   0, H_ - 1);   // tile row r holds x[clamp(y0-4+r)]
        int gc  = clampi(x0 - 4 + col, 0, W_ - 1);
        const float* ga = xb + ((size_t)c * H_ + gr) * W_ + gc;
        async_copy_b32(lds_off(&xs[c][r][col]), ga);
    }
    wait_async_all();
    __syncthreads();

    // ---- dilation: sum over channels of 5x5 max-plus, into LDS ----
    if (interior) {
        // Fast path: register-blocked 6-wide strips, 216 strips = 36 rows x 6.
        // Shares a 10-column register window across 6 outputs (75 -> 25 LDS
        // reads per element) and exposes per-row independent max chains.
        if (t < DT * 6) {
            const int ly  = t / 6;             // 0..35
            const int lx0 = (t - ly * 6) * 6;  // 0,6,...,30
            float acc[6] = {0.f, 0.f, 0.f, 0.f, 0.f, 0.f};
            #pragma unroll
            for (int c = 0; c < CIN; ++c) {
                float m[6] = {-INFINITY, -INFINITY, -INFINITY, -INFINITY, -INFINITY, -INFINITY};
                #pragma unroll
                for (int di = 0; di < K_; ++di) {
                    float r[10];
                    #pragma unroll
                    for (int k = 0; k < 10; ++k) r[k] = xs[c][ly + di][lx0 + k];
                    float w0 = wds[(c * K_ + di) * K_ + 0];
                    float w1 = wds[(c * K_ + di) * K_ + 1];
                    float w2 = wds[(c * K_ + di) * K_ + 2];
                    float w3 = wds[(c * K_ + di) * K_ + 3];
                    float w4 = wds[(c * K_ + di) * K_ + 4];
                    #pragma unroll
                    for (int j = 0; j < 6; ++j) {
                        float v0 = fmaxf(r[j]     + w0, r[j + 1] + w1);
                        float v1 = fmaxf(r[j + 2] + w2, r[j + 3] + w3);
                        float v  = fmaxf(fmaxf(v0, v1), r[j + 4] + w4);
                        m[j] = fmaxf(m[j], v);
                    }
                }
                #pragma unroll
                for (int j = 0; j < 6; ++j) acc[j] += m[j];
            }
            #pragma unroll
            for (int j = 0; j < 6; ++j) dil[ly][lx0 + j] = acc[j];
        }
    } else {
        // Border path: fully general double-clamp (erosion-level clamp first,
        // dilation-window clamp baked into the x-tile load).
        for (int e = t; e < DT * DT; e += 256) {
            int ly = e / DT, lx = e - ly * DT;
            int cy = clampi(y0 - 2 + ly, 0, H_ - 1);
            int cx = clampi(x0 - 2 + lx, 0, W_ - 1);
            int ry = cy - y0 + 4;              // 2..37
            int rx = cx - x0 + 4;
            float acc = 0.0f;
            #pragma unroll
            for (int c = 0; c < CIN; ++c) {
                float m = -INFINITY;
                #pragma unroll
                for (int di = 0; di < K_; ++di) {
                    float v0 = fmaxf(xs[c][ry + di - 2][rx - 2] + wds[(c * K_ + di) * K_ + 0],
                                     xs[c][ry + di - 2][rx - 1] + wds[(c * K_ + di) * K_ + 1]);
                    float v1 = fmaxf(xs[c][ry + di - 2][rx    ] + wds[(c * K_ + di) * K_ + 2],
                                     xs[c][ry + di - 2][rx + 1] + wds[(c * K_ + di) * K_ + 3]);
                    float v  = fmaxf(fmaxf(v0, v1),
                                     xs[c][ry + di - 2][rx + 2] + wds[(c * K_ + di) * K_ + 4]);
                    m = fmaxf(m, v);
                }
                acc += m;
            }
            dil[ly][lx] = acc;
        }
    }
    __syncthreads();

    // ---- erosion: 4-wide strips, b128 LDS reads, one b128 NT store ----
    {
        const int oy  = t >> 3;          // 0..31
        const int ox0 = (t & 7) * 4;     // 0..28
        float mn0 = INFINITY, mn1 = INFINITY, mn2 = INFINITY, mn3 = INFINITY;
        #pragma unroll
        for (int di = 0; di < K_; ++di) {
            v4f a = *(const v4f*)&dil[oy + di][ox0];
            v4f bb = *(const v4f*)&dil[oy + di][ox0 + 4];
            float r[8] = {a.x, a.y, a.z, a.w, bb.x, bb.y, bb.z, bb.w};
            #pragma unroll
            for (int dj = 0; dj < K_; ++dj) {
                float w = wes[di * K_ + dj];
                mn0 = fminf(mn0, r[dj]     - w);
                mn1 = fminf(mn1, r[dj + 1] - w);
                mn2 = fminf(mn2, r[dj + 2] - w);
                mn3 = fminf(mn3, r[dj + 3] - w);
            }
        }
        v4f res;
        res.x = mn0; res.y = mn1; res.z = mn2; res.w = mn3;
        float* ob = out + (((size_t)b * COUT + o) * H_ + (y0 + oy)) * W_ + (x0 + ox0);
        __builtin_nontemporal_store(res, (v4f*)ob);
    }
}

extern "C" void kernel_launch(void* const* d_in, const int* in_sizes, int n_in,
                              void* d_out, int out_size, void* d_ws, size_t ws_size,
                              hipStream_t stream) {
    (void)in_sizes; (void)n_in; (void)out_size; (void)d_ws; (void)ws_size;
    const float* x  = (const float*)d_in[0];   // (4,3,256,256)
    const float* wd = (const float*)d_in[1];   // (16,3,5,5)
    const float* we = (const float*)d_in[2];   // (16,5,5)
    float* out = (float*)d_out;                // (4,16,256,256)

    dim3 grid((H_ / TILE) * (W_ / TILE), COUT, B_);  // 64 x 16 x 4 = 4096 blocks
    dim3 block(256);                                  // 8 wave32s
    closing_fused_kernel<<<grid, block, 0, stream>>>(x, wd, we, out);
}